// WassersteinLoss_67808943669945
// MI455X (gfx1250) — compile-verified
//
#include <hip/hip_runtime.h>
#include <hip/hip_bf16.h>
#include <stdint.h>

#define TILE_ROWS 256
#define TILE_F    (TILE_ROWS * 5)   // 1280 floats per array per tile
#define THREADS   256
#define DEG2RAD_F (3.1415926f / 180.0f)

typedef __attribute__((ext_vector_type(2))) float        v2f;
typedef __attribute__((ext_vector_type(8))) float        v8f;
typedef __attribute__((ext_vector_type(4))) unsigned int u32x4;
typedef __attribute__((ext_vector_type(8))) int          i32x8;
typedef __attribute__((ext_vector_type(4))) int          i32x4;

#ifdef __has_builtin
#if __has_builtin(__builtin_amdgcn_tensor_load_to_lds) && __has_builtin(__builtin_amdgcn_s_wait_tensorcnt)
#define USE_TDM 1
#endif
#if __has_builtin(__builtin_amdgcn_wmma_f32_16x16x4_f32)
#define USE_WMMA_RED 1
#endif
#endif

// ---------------------------------------------------------------------------
// Wave32 sum via FP32 WMMA: A(16x4) holds per-lane x in K=0 slot, B = ones.
// D[m][n] = x[m] + x[m+16]; lane sums its 8 D VGPRs (rows m = v or v+8), then
// one xor-16 shuffle combines the two half-wave partitions. Full f32 precision.
// EXEC must be all ones at call sites (it is: called from convergent code).
// ---------------------------------------------------------------------------
__device__ __forceinline__ float wave_sum32(float x) {
#if defined(USE_WMMA_RED)
    v2f a; a[0] = x;    a[1] = 0.0f;
    v2f b; b[0] = 1.0f; b[1] = 1.0f;
    v8f c = {0.f, 0.f, 0.f, 0.f, 0.f, 0.f, 0.f, 0.f};
    v8f d = __builtin_amdgcn_wmma_f32_16x16x4_f32(false, a, false, b,
                                                  (short)0, c, false, false);
    float s = ((d[0] + d[1]) + (d[2] + d[3])) + ((d[4] + d[5]) + (d[6] + d[7]));
    s += __shfl_xor(s, 16, 32);
    return s;
#else
    for (int off = 16; off > 0; off >>= 1) x += __shfl_xor(x, off, 32);
    return x;
#endif
}

#if defined(USE_TDM)
// 1D tile copy global -> LDS via Tensor Data Mover (ISA 08, D# groups 0..3).
// nelem <= 65535 f32 elements. All descriptor words forced wave-uniform.
// This toolchain exposes the 6-arg builtin:
//   (uint32x4 g0, int32x8 g1, int32x4 g2, int32x4 g3, int32x8 g4, i32 cpol)
__device__ __forceinline__ void tdm_load_1d(const float* gsrc, unsigned lds_off,
                                            unsigned nelem) {
    unsigned long long ga = (unsigned long long)(uintptr_t)gsrc;
    unsigned galo = (unsigned)__builtin_amdgcn_readfirstlane((int)(unsigned)ga);
    unsigned gahi = (unsigned)__builtin_amdgcn_readfirstlane((int)(unsigned)(ga >> 32));
    unsigned loff = (unsigned)__builtin_amdgcn_readfirstlane((int)lds_off);
    unsigned ne   = (unsigned)__builtin_amdgcn_readfirstlane((int)nelem);

    u32x4 g0;
    g0[0] = 1u;                                   // count=1 (valid), user mode
    g0[1] = loff;                                 // lds_addr (bytes)
    g0[2] = galo;                                 // global_addr[31:0]
    g0[3] = (gahi & 0x1FFFFFFu) | (2u << 30);     // global_addr[56:32] | type=2

    i32x8 g1;
    g1[0] = (int)(2u << 16);                      // data_size=2 -> 4 bytes
    g1[1] = (int)((ne & 0xFFFFu) << 16);          // tensor_dim0[15:0]
    g1[2] = (int)(((ne >> 16) & 0xFFFFu) | (1u << 16)); // tensor_dim0 hi | tensor_dim1=1
    g1[3] = (int)((ne & 0xFFFFu) << 16);          // tensor_dim1 hi=0 | tile_dim0=ne
    g1[4] = 1;                                    // tile_dim1=1, tile_dim2=0
    g1[5] = (int)ne;                              // tensor_dim0_stride lo32
    g1[6] = 0;                                    // stride hi, dim1_stride lo
    g1[7] = 0;

    i32x4 gz; gz[0] = 0; gz[1] = 0; gz[2] = 0; gz[3] = 0;
    i32x8 gz8; gz8[0] = 0; gz8[1] = 0; gz8[2] = 0; gz8[3] = 0;
    gz8[4] = 0; gz8[5] = 0; gz8[6] = 0; gz8[7] = 0;
    __builtin_amdgcn_tensor_load_to_lds(g0, g1, gz, gz, gz8, 0);
}
#endif

// Per-row GWD loss. base: buf region with pred | target | weight tiles.
__device__ __forceinline__ float row_loss(const float* base, int lr) {
    const float* P = base + lr * 5;
    const float* Q = base + TILE_F + lr * 5;
    const float* W = base + 2 * TILE_F + lr * 5;

    float u0 = W[0], u1 = W[1], u2 = W[2], u3 = W[3], u4 = W[4];
    bool m = (u0 != 0.f) | (u1 != 0.f) | (u2 != 0.f) | (u3 != 0.f) | (u4 != 0.f);

    float cx1 = P[0], cy1 = P[1], ww1 = P[2], hh1 = P[3], th1 = P[4];
    float cx2 = Q[0], cy2 = Q[1], ww2 = Q[2], hh2 = Q[3], th2 = Q[4];

    float t1 = th1 * DEG2RAD_F, t2 = th2 * DEG2RAD_F;
    float c1 = __cosf(t1), s1 = __sinf(t1);
    float c2 = __cosf(t2), s2 = __sinf(t2);
    float W1 = ww1 * ww1, H1 = hh1 * hh1;
    float W2 = ww2 * ww2, H2 = hh2 * hh2;

    float a1 = (W1 * c1 * c1 + H1 * s1 * s1) * 0.25f;
    float b1 = ((W1 - H1) * s1 * c1) * 0.25f;
    float d1 = (W1 * s1 * s1 + H1 * c1 * c1) * 0.25f;
    float a2 = (W2 * c2 * c2 + H2 * s2 * s2) * 0.25f;
    float b2 = ((W2 - H2) * s2 * c2) * 0.25f;
    float d2 = (W2 * s2 * s2 + H2 * c2 * c2) * 0.25f;

    float dx = cx1 - cx2, dy = cy1 - cy2;
    float item1 = dx * dx + dy * dy;

    // s1 = sqrtm(cov1) = (cov1 + sqrt(det)*I) / sqrt(tr + 2*sqrt(det))
    float det1 = fmaxf(a1 * d1 - b1 * b1, 0.f);
    float sd = __builtin_amdgcn_sqrtf(det1);
    float tt = __builtin_amdgcn_sqrtf(fmaxf(a1 + d1 + 2.f * sd, 1e-12f));
    float it = __builtin_amdgcn_rcpf(tt);
    float sp = (a1 + sd) * it, sq = b1 * it, sr = (d1 + sd) * it;

    // M = s1 * cov2 * s1
    float T00 = sp * a2 + sq * b2, T01 = sp * b2 + sq * d2;
    float T10 = sq * a2 + sr * b2, T11 = sq * b2 + sr * d2;
    float M00 = T00 * sp + T01 * sq, M01 = T00 * sq + T01 * sr;
    float M10 = T10 * sp + T11 * sq, M11 = T10 * sq + T11 * sr;

    // trace(sqrtm(M)) = sqrt(clamp(trM + 2*sqrt(clamp(detM,0)), 1e-12))
    float detM = fmaxf(M00 * M11 - M01 * M10, 0.f);
    float sdM = __builtin_amdgcn_sqrtf(detM);
    float tM = __builtin_amdgcn_sqrtf(fmaxf(M00 + M11 + 2.f * sdM, 1e-12f));

    float item2 = (a1 + d1 + a2 + d2) - 2.f * tM;
    float v = fminf(fmaxf(item1 + item2 + 1e-8f, 0.f), 1e6f);
    float dist = __builtin_amdgcn_sqrtf(v);
    float l = 1.f - __builtin_amdgcn_rcpf(dist + 2.f);
    return m ? l : 0.f;
}

__global__ __launch_bounds__(THREADS)
void gwd_partial_kernel(const float* __restrict__ pred,
                        const float* __restrict__ target,
                        const float* __restrict__ weight,
                        float* __restrict__ partials, int N, int numTiles) {
    __shared__ float lds[2 * 3 * TILE_F];   // double-buffered pred|target|weight
    __shared__ float wsum[THREADS / 32];
    const int tid = (int)threadIdx.x;
    float acc = 0.f;

#if defined(USE_TDM)
    const int wv = __builtin_amdgcn_readfirstlane(tid) >> 5;  // uniform wave id
    // prime: waves 0/1/2 each DMA one array of tile blockIdx.x into buffer 0
    if (wv < 3) {
        int t0 = (int)blockIdx.x;
        int rows = N - t0 * TILE_ROWS; rows = rows > TILE_ROWS ? TILE_ROWS : rows;
        const float* src = (wv == 0) ? pred : ((wv == 1) ? target : weight);
        tdm_load_1d(src + (size_t)t0 * TILE_F,
                    (unsigned)(uintptr_t)&lds[wv * TILE_F], (unsigned)(rows * 5));
    }
    int i = 0;
    for (int t = (int)blockIdx.x; t < numTiles; t += (int)gridDim.x, ++i) {
        int buf = i & 1;
        int tn = t + (int)gridDim.x;
        if (wv < 3) {
            if (tn < numTiles) {
                int rows = N - tn * TILE_ROWS; rows = rows > TILE_ROWS ? TILE_ROWS : rows;
                const float* src = (wv == 0) ? pred : ((wv == 1) ? target : weight);
                tdm_load_1d(src + (size_t)tn * TILE_F,
                            (unsigned)(uintptr_t)&lds[((buf ^ 1) * 3 + wv) * TILE_F],
                            (unsigned)(rows * 5));
                __builtin_amdgcn_s_wait_tensorcnt(1);   // current tile landed
            } else {
                __builtin_amdgcn_s_wait_tensorcnt(0);   // drain
            }
        }
        __syncthreads();
        int row = t * TILE_ROWS + tid;
        if (row < N) acc += row_loss(&lds[buf * 3 * TILE_F], tid);
        __syncthreads();                                // buffer reuse fence
    }
#else
    for (int t = (int)blockIdx.x; t < numTiles; t += (int)gridDim.x) {
        int rows = N - t * TILE_ROWS; rows = rows > TILE_ROWS ? TILE_ROWS : rows;
        int len = rows * 5;
        for (int j = tid; j < len; j += THREADS) {
            lds[0 * TILE_F + j] = pred[(size_t)t * TILE_F + j];
            lds[1 * TILE_F + j] = target[(size_t)t * TILE_F + j];
            lds[2 * TILE_F + j] = weight[(size_t)t * TILE_F + j];
        }
        __syncthreads();
        int row = t * TILE_ROWS + tid;
        if (row < N) acc += row_loss(&lds[0], tid);
        __syncthreads();
    }
#endif

    float s = wave_sum32(acc);
    if ((tid & 31) == 0) wsum[tid >> 5] = s;
    __syncthreads();
    if (tid == 0) {
        float tot = 0.f;
#pragma unroll
        for (int k = 0; k < THREADS / 32; ++k) tot += wsum[k];
        partials[blockIdx.x] = tot;
    }
}

__global__ __launch_bounds__(THREADS)
void gwd_finalize_kernel(const float* __restrict__ partials, int n,
                         const int* __restrict__ avgf, float* __restrict__ out) {
    __shared__ float wsum[THREADS / 32];
    const int tid = (int)threadIdx.x;
    float acc = 0.f;
    for (int i = tid; i < n; i += THREADS) acc += partials[i];
    float s = wave_sum32(acc);
    if ((tid & 31) == 0) wsum[tid >> 5] = s;
    __syncthreads();
    if (tid == 0) {
        float tot = 0.f;
#pragma unroll
        for (int k = 0; k < THREADS / 32; ++k) tot += wsum[k];
        out[0] = tot / (float)(*avgf);
    }
}

extern "C" void kernel_launch(void* const* d_in, const int* in_sizes, int n_in,
                              void* d_out, int out_size, void* d_ws, size_t ws_size,
                              hipStream_t stream) {
    const float* pred   = (const float*)d_in[0];
    const float* target = (const float*)d_in[1];
    const float* weight = (const float*)d_in[2];
    const int*   avgf   = (const int*)d_in[3];
    float* out = (float*)d_out;

    int N = in_sizes[0] / 5;
    int numTiles = (N + TILE_ROWS - 1) / TILE_ROWS;

    int maxBlocks = 4096;
    size_t wsCap = ws_size / sizeof(float);
    if (wsCap < (size_t)maxBlocks) maxBlocks = (int)wsCap;
    if (maxBlocks < 1) maxBlocks = 1;
    int blocks = numTiles < maxBlocks ? numTiles : maxBlocks;
    if (blocks < 1) blocks = 1;

    float* partials = (float*)d_ws;
    gwd_partial_kernel<<<blocks, THREADS, 0, stream>>>(pred, target, weight,
                                                       partials, N, numTiles);
    gwd_finalize_kernel<<<1, THREADS, 0, stream>>>(partials, blocks, avgf, out);
}